// VanillaUpsampler_16398185136583
// MI455X (gfx1250) — compile-verified
//
#include <hip/hip_runtime.h>

// Problem constants (static in the reference)
#define B_ 8
#define N_ 512
#define T_ 8192
#define D_ 256

typedef float v2f __attribute__((ext_vector_type(2)));
typedef float v8f __attribute__((ext_vector_type(8)));

// ---------------------------------------------------------------------------
// Kernel 1: per-batch inclusive cumsum of durations -> segment end frames e[b,n]
// Tiny (8 x 512 serial adds); one thread per batch row.
// ---------------------------------------------------------------------------
__global__ void vu_cumsum_kernel(const int* __restrict__ dur, int* __restrict__ e) {
    int b = threadIdx.x;
    if (b < B_) {
        int acc = 0;
        for (int n = 0; n < N_; ++n) {
            acc += dur[b * N_ + n];
            e[b * N_ + n] = acc;
        }
    }
}

// Branchless lower-bound: smallest n with e[n] > t  (count of e[n] <= t).
// N_ = 512 is a power of two; e[N_-1] == T_ > t guarantees result < N_.
__device__ __forceinline__ int vu_find_seg(const int* __restrict__ eb, int t) {
    int pos = 0;
#pragma unroll
    for (int s = N_ / 2; s > 0; s >>= 1) {
        if (eb[pos + s - 1] <= t) pos += s;
    }
    return pos;
}

// ---------------------------------------------------------------------------
// Kernel 2: positions[b,t] = t - start(seg(b,t))        (B*T threads)
// ---------------------------------------------------------------------------
__global__ void vu_positions_kernel(const int* __restrict__ dur,
                                    const int* __restrict__ e,
                                    float* __restrict__ pos) {
    int id = blockIdx.x * blockDim.x + threadIdx.x;   // B_*T_ exactly
    int b = id / T_;
    int t = id - b * T_;
    const int* eb = e + b * N_;
    int n = vu_find_seg(eb, t);
    int start = eb[n] - dur[b * N_ + n];
    pos[id] = (float)t - (float)start;
}

// ---------------------------------------------------------------------------
// Kernel 3: weights[b,n,t] via the reference's exact predicate, float4 stores.
// One thread per (b,n, 4 frames). Pure store bandwidth (134 MB).
// ---------------------------------------------------------------------------
__global__ void vu_weights_kernel(const int* __restrict__ dur,
                                  const int* __restrict__ e,
                                  float* __restrict__ w) {
    int id = blockIdx.x * blockDim.x + threadIdx.x;   // B_*N_*T_/4 exactly
    int bn = id / (T_ / 4);
    int t0 = (id - bn * (T_ / 4)) * 4;
    float ef = (float)e[bn];
    float df = (float)dur[bn];
    float c  = ef - 0.5f * df;                        // segment center (exact in f32)
    float hd = 0.5f * df;
    float4 o;
    o.x = (fabsf((float)(t0 + 0) - c + 1e-6f) < hd) ? 1.0f : 0.0f;
    o.y = (fabsf((float)(t0 + 1) - c + 1e-6f) < hd) ? 1.0f : 0.0f;
    o.z = (fabsf((float)(t0 + 2) - c + 1e-6f) < hd) ? 1.0f : 0.0f;
    o.w = (fabsf((float)(t0 + 3) - c + 1e-6f) < hd) ? 1.0f : 0.0f;
    *(float4*)(w + (size_t)bn * T_ + t0) = o;
}

// ---------------------------------------------------------------------------
// Kernel 4: out[b,t,:] = sum_n weights[b,n,t] * x[b,n,:] via V_WMMA_F32_16X16X4_F32.
// One wave32 per (b, 16-frame tile). Since min duration >= 8 (plus at most one
// short final segment), a 16-frame tile touches <= 4 consecutive segments
// n0..n0+3 -> exact K=4 WMMA shape. A is the one-hot selection (16x4), B is
// x[n0..n0+3, d0:d0+16] (4x16), 16 WMMAs sweep D=256. f32 WMMA with one-hot A
// is bit-exact vs the gather. EXEC is all-1s: grids divide exactly, no returns.
// ---------------------------------------------------------------------------
__global__ void vu_upsample_wmma_kernel(const float* __restrict__ x,
                                        const int* __restrict__ e,
                                        float* __restrict__ out) {
    const int lane = threadIdx.x & 31;
    const int wave = threadIdx.x >> 5;
    const int tile = blockIdx.x * (blockDim.x >> 5) + wave;  // B_*(T_/16) tiles
    const int b  = tile / (T_ / 16);
    const int tt = tile - b * (T_ / 16);
    const int t0 = tt * 16;
    const int m  = lane & 15;          // A: row M / B: col N / C: col N
    const int hi = lane >> 4;          // lane half selects K pair / C row offset

    const int* eb = e + b * N_;
    int myseg = vu_find_seg(eb, t0 + m);                 // segment of frame t0+m
    int n0 = __builtin_amdgcn_readfirstlane(myseg);      // segment of first frame

    // A-matrix (16x4 f32): VGPR0 -> K = 2*hi, VGPR1 -> K = 2*hi+1 (ISA 7.12.2)
    const int kb = hi * 2;
    v2f a;
    a.x = (myseg == n0 + kb)     ? 1.0f : 0.0f;
    a.y = (myseg == n0 + kb + 1) ? 1.0f : 0.0f;

    // B-matrix rows (clamped: rows past N-1 have a==0, load anything valid)
    int r0 = n0 + kb;     if (r0 > N_ - 1) r0 = N_ - 1;
    int r1 = n0 + kb + 1; if (r1 > N_ - 1) r1 = N_ - 1;
    const float* xb   = x + (size_t)b * N_ * D_;
    const float* row0 = xb + (size_t)r0 * D_;
    const float* row1 = xb + (size_t)r1 * D_;

    float* ob = out + ((size_t)b * T_ + t0) * D_;

    for (int dt = 0; dt < D_; dt += 16) {
        v2f bv;                            // B 4x16: VGPR0 -> K=2*hi, VGPR1 -> K=2*hi+1
        bv.x = row0[dt + m];
        bv.y = row1[dt + m];
        v8f cc = {};
        // 8 args: (neg_a, A, neg_b, B, c_mod, C, reuse_a, reuse_b)
        cc = __builtin_amdgcn_wmma_f32_16x16x4_f32(
                false, a, false, bv, (short)0, cc, false, false);
        // C/D layout: VGPR r -> frame t0 + r + 8*hi, lane%16 -> column dt+m
#pragma unroll
        for (int r = 0; r < 8; ++r) {
            ob[(size_t)(r + hi * 8) * D_ + dt + m] = cc[r];
        }
    }
}

// ---------------------------------------------------------------------------
// Launch: outputs concatenated as (positions[B,T], out[B,T,D], weights[B,N,T])
// ---------------------------------------------------------------------------
extern "C" void kernel_launch(void* const* d_in, const int* in_sizes, int n_in,
                              void* d_out, int out_size, void* d_ws, size_t ws_size,
                              hipStream_t stream) {
    const float* x  = (const float*)d_in[0];
    const int* dur  = (const int*)d_in[1];

    float* pos = (float*)d_out;                              // B*T
    float* big = pos + (size_t)B_ * T_;                      // B*T*D
    float* w   = big + (size_t)B_ * T_ * D_;                 // B*N*T

    int* e = (int*)d_ws;                                     // B*N ints (16 KB)

    vu_cumsum_kernel<<<1, 32, 0, stream>>>(dur, e);

    vu_positions_kernel<<<(B_ * T_) / 256, 256, 0, stream>>>(dur, e, pos);

    vu_weights_kernel<<<(B_ * N_ * (T_ / 4)) / 256, 256, 0, stream>>>(dur, e, w);

    // 4096 tiles, 8 waves (256 threads) per block -> 512 blocks, exact fit
    vu_upsample_wmma_kernel<<<(B_ * (T_ / 16)) / 8, 256, 0, stream>>>(x, e, big);
}